// GeneralAttention_29205777613488
// MI455X (gfx1250) — compile-verified
//
#include <hip/hip_runtime.h>

#define BDIM 16
#define LDIM 4096
#define HDIM 512
#define NEGV (-1e9f)

typedef __attribute__((ext_vector_type(2))) float v2f;
typedef __attribute__((ext_vector_type(8))) float v8f;

// ---------------------------------------------------------------------------
// Kernel 0: zero the accumulation buffers (colsum S in ws, context in d_out).
// ---------------------------------------------------------------------------
__global__ void k_zero(float* __restrict__ s, float* __restrict__ ctx) {
    int i = blockIdx.x * blockDim.x + threadIdx.x;   // 0 .. B*H-1
    s[i]   = 0.0f;
    ctx[i] = 0.0f;
}

// ---------------------------------------------------------------------------
// Kernel 1: S[b,d] = sum_l x[b,l,d].  Streaming pass 1 over 128MB (HBM).
// Block = 128 threads, each owns 4 consecutive d's (b128 loads); 128 lanes *
// 16B = one full 2KB row per iteration, 64 rows per block.
// ---------------------------------------------------------------------------
__global__ void k_colsum(const float* __restrict__ x, float* __restrict__ s) {
    const int b  = blockIdx.y;
    const int l0 = blockIdx.x * 64;
    const int d  = threadIdx.x * 4;
    const float* p = x + (size_t)b * LDIM * HDIM + (size_t)l0 * HDIM + d;
    float a0 = 0.0f, a1 = 0.0f, a2 = 0.0f, a3 = 0.0f;
#pragma unroll 4
    for (int r = 0; r < 64; ++r) {
        float4 v = *(const float4*)p;
        a0 += v.x; a1 += v.y; a2 += v.z; a3 += v.w;
        p += HDIM;
    }
    atomicAdd(&s[b * HDIM + d],     a0);
    atomicAdd(&s[b * HDIM + d + 1], a1);
    atomicAdd(&s[b * HDIM + d + 2], a2);
    atomicAdd(&s[b * HDIM + d + 3], a3);
}

// ---------------------------------------------------------------------------
// Kernel 2: wx[b,h] = sum_d S[b,d] * W[h,d]  ==  D = A(16xK) * B(Kx16) tiles
// via V_WMMA_F32_16X16X4_F32. One wave per 16-wide N-tile of H (32 blocks).
// A frag: lane%16 = M, VGPR j holds K = k0 + (lane/16)*2 + j.
// B frag: lane%16 = N, VGPR j holds K = k0 + (lane/16)*2 + j (B[k][n]=W[n][k]).
// C/D:    lanes 0-15 -> (M=r, N=lane); lanes 16-31 -> (M=r+8, N=lane-16).
// ---------------------------------------------------------------------------
__global__ void k_wx_wmma(const float* __restrict__ s, const float* __restrict__ w,
                          float* __restrict__ wx) {
    const int lane = threadIdx.x;          // 0..31, full wave, EXEC all ones
    const int mi   = lane & 15;
    const int kh   = (lane >> 4) << 1;     // 0 or 2
    const int n    = blockIdx.x * 16 + mi; // output column in H
    const float* sp = s + mi * HDIM;       // A row (batch mi)
    const float* wp = w + (size_t)n * HDIM;// B column n == W row n
    v8f c = {};
    for (int k0 = 0; k0 < HDIM; k0 += 4) {
        v2f a, bm;
        a[0]  = sp[k0 + kh];
        a[1]  = sp[k0 + kh + 1];
        bm[0] = wp[k0 + kh];
        bm[1] = wp[k0 + kh + 1];
        c = __builtin_amdgcn_wmma_f32_16x16x4_f32(false, a, false, bm,
                                                  (short)0, c, false, false);
    }
    const int mb = (lane < 16) ? 0 : 8;
#pragma unroll
    for (int r = 0; r < 8; ++r) {
        wx[(mb + r) * HDIM + n] = c[r];
    }
}

// ---------------------------------------------------------------------------
// Kernel 3: scores[b,l] = x[b,l,:] . wx[b,:], masked. Streaming pass 2 (L2).
// 8 waves/block, each wave does 4 rows; per row: 4x float4 per lane (512 f32),
// wave reduction via shfl_xor. wx row staged in LDS once per block.
// ---------------------------------------------------------------------------
__global__ void k_scores(const float* __restrict__ x, const float* __restrict__ wx,
                         const int* __restrict__ lengths, float* __restrict__ scores) {
    __shared__ float wxs[HDIM];
    const int b = blockIdx.y;
    for (int i = threadIdx.x; i < HDIM; i += blockDim.x) wxs[i] = wx[b * HDIM + i];
    __syncthreads();

    const int wave = threadIdx.x >> 5;
    const int lane = threadIdx.x & 31;
    const int len  = lengths[b];
    const int lbase = blockIdx.x * 32 + wave * 4;

    for (int rr = 0; rr < 4; ++rr) {
        const int l = lbase + rr;
        const float* row = x + (size_t)b * LDIM * HDIM + (size_t)l * HDIM;
        float acc = 0.0f;
#pragma unroll
        for (int seg = 0; seg < 4; ++seg) {
            const int d = seg * 128 + lane * 4;
            float4 v = *(const float4*)(row + d);
            acc += v.x * wxs[d] + v.y * wxs[d + 1] + v.z * wxs[d + 2] + v.w * wxs[d + 3];
        }
#pragma unroll
        for (int off = 16; off > 0; off >>= 1) acc += __shfl_xor(acc, off, 32);
        if (lane == 0) scores[b * LDIM + l] = (l < len) ? acc : NEGV;
    }
}

// ---------------------------------------------------------------------------
// Kernel 4: attn[b,:] = softmax(scores[b,:]). One 1024-thread block per batch;
// each thread owns 4 l's. Two-level (wave shfl + LDS) max/sum reductions.
// Masked entries are -1e9 -> exp underflows to 0, matching jnp.where+softmax.
// ---------------------------------------------------------------------------
__global__ void k_softmax(const float* __restrict__ scores, float* __restrict__ attn) {
    __shared__ float red[32];
    __shared__ float bc[2];
    const int b   = blockIdx.x;
    const int tid = threadIdx.x;            // 0..1023
    const int wid = tid >> 5;
    const int lane = tid & 31;

    float4 v = *(const float4*)(scores + b * LDIM + tid * 4);

    float m = fmaxf(fmaxf(v.x, v.y), fmaxf(v.z, v.w));
#pragma unroll
    for (int off = 16; off > 0; off >>= 1) m = fmaxf(m, __shfl_xor(m, off, 32));
    if (lane == 0) red[wid] = m;
    __syncthreads();
    if (tid == 0) {
        float t = red[0];
        for (int i = 1; i < 32; ++i) t = fmaxf(t, red[i]);
        bc[0] = t;
    }
    __syncthreads();
    const float M = bc[0];

    float4 e;
    e.x = __expf(v.x - M);
    e.y = __expf(v.y - M);
    e.z = __expf(v.z - M);
    e.w = __expf(v.w - M);
    float ssum = e.x + e.y + e.z + e.w;
#pragma unroll
    for (int off = 16; off > 0; off >>= 1) ssum += __shfl_xor(ssum, off, 32);
    if (lane == 0) red[wid] = ssum;
    __syncthreads();
    if (tid == 0) {
        float t = 0.0f;
        for (int i = 0; i < 32; ++i) t += red[i];
        bc[1] = t;
    }
    __syncthreads();
    const float inv = 1.0f / bc[1];

    e.x *= inv; e.y *= inv; e.z *= inv; e.w *= inv;
    *(float4*)(attn + b * LDIM + tid * 4) = e;
}

// ---------------------------------------------------------------------------
// Kernel 5: context[b,h] = sum_l attn[b,l] * x[b,l,h]. Streaming pass 3 (L2).
// Same b128 layout as k_colsum; attn[b,l] is block-uniform (scalar-loadable).
// ---------------------------------------------------------------------------
__global__ void k_context(const float* __restrict__ x, const float* __restrict__ attn,
                          float* __restrict__ ctx) {
    const int b  = blockIdx.y;
    const int l0 = blockIdx.x * 64;
    const int d  = threadIdx.x * 4;
    const float* p  = x + (size_t)b * LDIM * HDIM + (size_t)l0 * HDIM + d;
    const float* ap = attn + b * LDIM + l0;
    float a0 = 0.0f, a1 = 0.0f, a2 = 0.0f, a3 = 0.0f;
#pragma unroll 4
    for (int r = 0; r < 64; ++r) {
        const float a = ap[r];
        float4 v = *(const float4*)p;
        a0 += a * v.x; a1 += a * v.y; a2 += a * v.z; a3 += a * v.w;
        p += HDIM;
    }
    atomicAdd(&ctx[b * HDIM + d],     a0);
    atomicAdd(&ctx[b * HDIM + d + 1], a1);
    atomicAdd(&ctx[b * HDIM + d + 2], a2);
    atomicAdd(&ctx[b * HDIM + d + 3], a3);
}

// ---------------------------------------------------------------------------
extern "C" void kernel_launch(void* const* d_in, const int* in_sizes, int n_in,
                              void* d_out, int out_size, void* d_ws, size_t ws_size,
                              hipStream_t stream) {
    (void)in_sizes; (void)n_in; (void)out_size; (void)ws_size;

    const float* x       = (const float*)d_in[0];   // [B,L,H] f32
    const int*   lengths = (const int*)d_in[1];     // [B]
    const float* W       = (const float*)d_in[2];   // [H,H] f32

    float* out  = (float*)d_out;
    float* ctx  = out;                 // [B,H]  first output
    float* attn = out + BDIM * HDIM;   // [B,L]  second output

    float* ws     = (float*)d_ws;
    float* S      = ws;                      // [B,H] column sums
    float* wx     = ws + BDIM * HDIM;        // [B,H] projected sums
    float* scores = ws + 2 * BDIM * HDIM;    // [B,L]

    k_zero<<<(BDIM * HDIM) / 256, 256, 0, stream>>>(S, ctx);
    k_colsum<<<dim3(LDIM / 64, BDIM), 128, 0, stream>>>(x, S);
    k_wx_wmma<<<HDIM / 16, 32, 0, stream>>>(S, W, wx);
    k_scores<<<dim3(LDIM / 32, BDIM), 256, 0, stream>>>(x, wx, lengths, scores);
    k_softmax<<<BDIM, 1024, 0, stream>>>(scores, attn);
    k_context<<<dim3(LDIM / 64, BDIM), 128, 0, stream>>>(x, attn, ctx);
}